// CyclicConsistencyLoss_35888746726000
// MI455X (gfx1250) — compile-verified
//
#include <hip/hip_runtime.h>

typedef float v2f __attribute__((ext_vector_type(2)));
typedef float v4f __attribute__((ext_vector_type(4)));
typedef float v8f __attribute__((ext_vector_type(8)));
// 4-byte-aligned float pair for (possibly odd-offset) corner-pair gathers.
typedef float p2f __attribute__((ext_vector_type(2), aligned(4)));

// xor-16 exchange+add via a single DS_SWIZZLE (group-of-32: xor=0x10, and=0x1f).
__device__ __forceinline__ float xor16_add(float s) {
    const int sw = __builtin_amdgcn_ds_swizzle(__float_as_int(s), 0x401F);
    return s + __int_as_float(sw);
}

// Full-wave (32-lane) sum using V_WMMA_F32_16X16X4_F32 as the cross-lane
// reduction engine.
//   A (16x4, v2f/lane): A[m][0] = p(lane m), A[m][2] = p(lane m+16), rest 0
//   B (4x16)          : all ones (layout-independent)
//   D[m][n] = p(m) + p(m+16)
// Per-lane sum of the 8 D VGPRs yields rows 0..7 (lanes<16) / 8..15
// (lanes>=16); one swizzle-xor-16 add completes the 32-lane total.
// Requires EXEC == all ones (callers are fully uniform here).
__device__ __forceinline__ float wave_sum_wmma(float p) {
    v2f a; a.x = p;    a.y = 0.0f;
    v2f b; b.x = 1.0f; b.y = 1.0f;
    v8f c = {};
    v8f d = __builtin_amdgcn_wmma_f32_16x16x4_f32(
        false, a, false, b, (short)0, c, false, false);
    float s = ((d[0] + d[1]) + (d[2] + d[3])) + ((d[4] + d[5]) + (d[6] + d[7]));
    return xor16_add(s);
}

// One thread = 4 consecutive x-pixels of one row of one batch image.
// forward_disp is streamed once with non-temporal 128-bit loads (keep L2 for
// the backward_disp gathers). Gathers are paired 64-bit loads of adjacent
// corners with a border select. Batch index is wave-uniform (blocks never
// straddle an image) -> readfirstlane puts all plane bases in SGPRs so every
// gather is SGPR-base + 32-bit-offset addressing.
__global__ __launch_bounds__(256)
void warp_residual_kernel(const float* __restrict__ fwd,
                          const float* __restrict__ bwd,
                          float* __restrict__ partials,
                          int H, int W) {
    const int tid         = blockIdx.x * blockDim.x + threadIdx.x;
    const int quadsPerRow = W >> 2;
    const int quadsPerImg = H * quadsPerRow;   // 65536; divisible by 256
    const int b  = __builtin_amdgcn_readfirstlane(tid / quadsPerImg);
    const int r  = tid - b * quadsPerImg;
    const int y  = r / quadsPerRow;
    const int xq = (r - y * quadsPerRow) << 2;

    const size_t plane = (size_t)H * (size_t)W;
    const float* __restrict__ fdx = fwd + (size_t)(2 * b)     * plane;
    const float* __restrict__ fdy = fwd + (size_t)(2 * b + 1) * plane;
    const float* __restrict__ bdx = bwd + (size_t)(2 * b)     * plane;
    const float* __restrict__ bdy = bwd + (size_t)(2 * b + 1) * plane;

    const unsigned rowOff = (unsigned)(y * W + xq);
    const v4f dx4 = __builtin_nontemporal_load((const v4f*)(fdx + rowOff));
    const v4f dy4 = __builtin_nontemporal_load((const v4f*)(fdy + rowOff));

    const float fy   = (float)y;
    const float wMax = (float)(W - 1);
    const float hMax = (float)(H - 1);

    float acc = 0.0f;
#pragma unroll
    for (int i = 0; i < 4; ++i) {
        const float fdx_i = dx4[i];
        const float fdy_i = dy4[i];

        const float gx = fminf(fmaxf((float)(xq + i) + fdx_i, 0.0f), wMax);
        const float gy = fminf(fmaxf(fy + fdy_i, 0.0f), hMax);

        const float x0f = floorf(gx);
        const float y0f = floorf(gy);
        const float wx  = gx - x0f;
        const float wy  = gy - y0f;

        const int x0 = (int)x0f;
        const int y0 = (int)y0f;
        const int y1 = min(y0 + 1, H - 1);
        // Pair base: always in-bounds; if x0 == W-1 then wx == 0 and the
        // pair's .y element IS pixel W-1, so selecting .y for v00 is exact.
        const int  xb = min(x0, W - 2);
        const bool hi = x0 > xb;

        const unsigned i0 = (unsigned)(y0 * W + xb);
        const unsigned i1 = (unsigned)(y1 * W + xb);

        const p2f px0 = *(const p2f*)(bdx + i0);  // dx plane, top corners
        const p2f px1 = *(const p2f*)(bdx + i1);  // dx plane, bottom corners
        const p2f py0 = *(const p2f*)(bdy + i0);  // dy plane, top corners
        const p2f py1 = *(const p2f*)(bdy + i1);  // dy plane, bottom corners

        const float v00x = hi ? px0.y : px0.x;
        const float v10x = hi ? px1.y : px1.x;
        const float v00y = hi ? py0.y : py0.x;
        const float v10y = hi ? py1.y : py1.x;

        const float topx = fmaf(wx, px0.y - v00x, v00x);
        const float botx = fmaf(wx, px1.y - v10x, v10x);
        const float topy = fmaf(wx, py0.y - v00y, v00y);
        const float boty = fmaf(wx, py1.y - v10y, v10y);

        const float wbx = fmaf(wy, botx - topx, topx);
        const float wby = fmaf(wy, boty - topy, topy);

        const float rx = fdx_i + wbx;
        const float ry = fdy_i + wby;
        acc = fmaf(rx, rx, acc);
        acc = fmaf(ry, ry, acc);
    }

    // Wave-level WMMA reduction (uniform control flow; EXEC all ones here).
    const float wsum = wave_sum_wmma(acc);

    __shared__ float lds[8];  // 256 threads / wave32 = 8 waves
    const int lane = threadIdx.x & 31;
    const int wid  = threadIdx.x >> 5;
    if (lane == 0) lds[wid] = wsum;
    __syncthreads();
    if (threadIdx.x == 0) {
        float s = 0.0f;
#pragma unroll
        for (int w = 0; w < 8; ++w) s += lds[w];
        partials[blockIdx.x] = s;  // deterministic: no float atomics
    }
}

// Single wave32 reduces all block partials (count divisible by 128 -> fully
// uniform loop, EXEC all ones at the WMMA) and applies 1/N.
__global__ __launch_bounds__(32)
void finalize_kernel(const float* __restrict__ partials,
                     float* __restrict__ out,
                     int nPartials, float invN) {
    const v4f* p4 = (const v4f*)partials;
    const int n4 = nPartials >> 2;
    float acc = 0.0f;
    for (int i = threadIdx.x; i < n4; i += 32) {
        const v4f v = p4[i];
        acc += (v[0] + v[1]) + (v[2] + v[3]);
    }
    const float tot = wave_sum_wmma(acc);
    if (threadIdx.x == 0) out[0] = tot * invN;
}

extern "C" void kernel_launch(void* const* d_in, const int* in_sizes, int n_in,
                              void* d_out, int out_size, void* d_ws, size_t ws_size,
                              hipStream_t stream) {
    const float* fwd = (const float*)d_in[0];
    const float* bwd = (const float*)d_in[1];
    float* out       = (float*)d_out;
    float* partials  = (float*)d_ws;

    const int H = 512, W = 512;                       // per reference setup
    const long long n = (long long)in_sizes[0];       // B * 2 * H * W
    const int B = (int)(n / (2LL * H * W));

    const int quads = B * H * (W / 4);                // 2,097,152
    const int block = 256;
    const int grid  = quads / block;                  // 8192 (exact)

    warp_residual_kernel<<<grid, block, 0, stream>>>(fwd, bwd, partials, H, W);

    const float invN = 1.0f / (float)n;
    finalize_kernel<<<1, 32, 0, stream>>>(partials, out, grid, invN);
}